// NNFlow_39367670235849
// MI455X (gfx1250) — compile-verified
//
#include <hip/hip_runtime.h>
#include <hip/hip_bf16.h>

// NNFlow: tiny-MLP normalizing flow, per-sample analytic Jacobian + log|det|.
// D=2, DH=5, NH=3, n=4.19M. Roofline: 24 B/sample -> ~4.3 us memory floor at
// 23.3 TB/s; ~225 FMA + 22 tanh + ~85 pointwise ops/sample -> VALU-bound.
// Strategy: lane-per-sample f32 VALU (wave32), hardware v_tanh_f32, async-DMA
// weight staging (gfx1250 ASYNCcnt path). WMMA rejected: for 5x5/5x2 matrices
// the per-layer B-operand repack (bpermute+cndmask) costs as many issue slots
// as the FMAs it would replace.

#define ALPHA_ 0.3f
#define BETA_  0.7f          // 1 - ALPHA
#define D_  2
#define DH_ 5
#define NH_ 3
#define NW_ (DH_*D_ + NH_*DH_*DH_ + D_*DH_)   // 10 + 75 + 10 = 95

__device__ __forceinline__ float tanh_fast(float x) {
#if defined(__gfx1250__) && __has_builtin(__builtin_amdgcn_tanhf)
    return __builtin_amdgcn_tanhf(x);          // V_TANH_F32 (CDNA5 trans op)
#elif defined(__gfx1250__) && __has_builtin(__builtin_amdgcn_tanh_f32)
    return __builtin_amdgcn_tanh_f32(x);
#else
    return tanhf(x);
#endif
}

template <int SPT, bool GUARD>
__global__ __launch_bounds__(256)
void nnflow_kernel(const float* __restrict__ xj,
                   const float* __restrict__ gW0,
                   const float* __restrict__ gWs,
                   const float* __restrict__ gWout,
                   float* __restrict__ out,
                   int n, int total_threads)
{
    __shared__ float wlds[NW_];

    // ---- Stage the 95 shared weight floats into LDS via CDNA5 async DMA ----
    // Branchless: clamp lane id, select source pointer with cndmask (no exec
    // branching); duplicate lanes rewrite wlds[NW_-1] with identical data.
    {
        int t = threadIdx.x;
        t = (t < NW_) ? t : (NW_ - 1);
        const float* gp = gW0 + t;
        if (t >= DH_*D_)                  gp = gWs   + (t - DH_*D_);
        if (t >= DH_*D_ + NH_*DH_*DH_)    gp = gWout + (t - DH_*D_ - NH_*DH_*DH_);
        unsigned long long ga = (unsigned long long)gp;
        unsigned           la = (unsigned)(unsigned long long)(&wlds[t]); // low 32b of flat LDS addr == LDS offset
        asm volatile("global_load_async_to_lds_b32 %0, %1, off"
                     :: "v"(la), "v"(ga) : "memory");
        asm volatile("s_wait_asynccnt 0" ::: "memory");
        __syncthreads();
    }

    // ---- Broadcast weights into registers (amortized over SPT samples) ----
    float W0[DH_][D_], Ws[NH_][DH_][DH_], Wout[D_][DH_];
    #pragma unroll
    for (int r = 0; r < DH_; ++r)
        #pragma unroll
        for (int c = 0; c < D_; ++c) W0[r][c] = wlds[r*D_ + c];
    #pragma unroll
    for (int l = 0; l < NH_; ++l)
        #pragma unroll
        for (int r = 0; r < DH_; ++r)
            #pragma unroll
            for (int c = 0; c < DH_; ++c)
                Ws[l][r][c] = wlds[DH_*D_ + (l*DH_ + r)*DH_ + c];
    #pragma unroll
    for (int r = 0; r < D_; ++r)
        #pragma unroll
        for (int c = 0; c < DH_; ++c)
            Wout[r][c] = wlds[DH_*D_ + NH_*DH_*DH_ + r*DH_ + c];

    const int tid = blockIdx.x * blockDim.x + threadIdx.x;
    const long long step = 3ll * total_threads;
    const float* px = xj  + 3ll * tid;
    float*       po = out + 3ll * tid;

    #pragma unroll 2
    for (int k = 0; k < SPT; ++k) {
        if (GUARD) {
            long long s = (long long)tid + (long long)k * total_threads;
            if (s >= n) break;
        }
        // next strided record (streaming; speculative prefetch is fault-free)
        __builtin_prefetch(px + step, 0, 1);

        const float x0 = px[0], x1 = px[1], lj = px[2];

        // ---- layer 0: z = W0 x ; J = diag(d) W0 ----
        float h[DH_], j0[DH_], j1[DH_];
        #pragma unroll
        for (int r = 0; r < DH_; ++r) {
            float z = fmaf(W0[r][0], x0, W0[r][1] * x1);
            float t = tanh_fast(z);
            float d = fmaf(-BETA_ * t, t, 1.0f);        // ALPHA + BETA*(1-t^2)
            h[r]  = fmaf(ALPHA_, z, BETA_ * t);
            j0[r] = d * W0[r][0];
            j1[r] = d * W0[r][1];
        }

        // ---- hidden layers: z = Ws h ; J = diag(d) (Ws J) ----
        #pragma unroll
        for (int l = 0; l < NH_; ++l) {
            float hn[DH_], a0[DH_], a1[DH_];
            #pragma unroll
            for (int r = 0; r < DH_; ++r) {
                float z = 0.f, a = 0.f, b = 0.f;
                #pragma unroll
                for (int c = 0; c < DH_; ++c) {
                    const float w = Ws[l][r][c];
                    z = fmaf(w, h[c],  z);
                    a = fmaf(w, j0[c], a);
                    b = fmaf(w, j1[c], b);
                }
                float t = tanh_fast(z);
                float d = fmaf(-BETA_ * t, t, 1.0f);
                hn[r] = fmaf(ALPHA_, z, BETA_ * t);
                a0[r] = d * a;
                a1[r] = d * b;
            }
            #pragma unroll
            for (int r = 0; r < DH_; ++r) { h[r] = hn[r]; j0[r] = a0[r]; j1[r] = a1[r]; }
        }

        // ---- output layer: y = act(Wout h) ; Jf = diag(d) (Wout J) (2x2) ----
        float y0, y1, Jf[D_][D_];
        #pragma unroll
        for (int r = 0; r < D_; ++r) {
            float z = 0.f, a = 0.f, b = 0.f;
            #pragma unroll
            for (int c = 0; c < DH_; ++c) {
                const float w = Wout[r][c];
                z = fmaf(w, h[c],  z);
                a = fmaf(w, j0[c], a);
                b = fmaf(w, j1[c], b);
            }
            float t = tanh_fast(z);
            float d = fmaf(-BETA_ * t, t, 1.0f);
            float y = fmaf(ALPHA_, z, BETA_ * t);
            if (r == 0) y0 = y; else y1 = y;
            Jf[r][0] = d * a;
            Jf[r][1] = d * b;
        }

        const float det    = fmaf(Jf[0][0], Jf[1][1], -Jf[0][1] * Jf[1][0]);
        const float logdet = __logf(fabsf(det));       // v_log_f32 path

        po[0] = y0;
        po[1] = y1;
        po[2] = lj + logdet;

        px += step;
        po += step;
    }
}

extern "C" void kernel_launch(void* const* d_in, const int* in_sizes, int n_in,
                              void* d_out, int out_size, void* d_ws, size_t ws_size,
                              hipStream_t stream)
{
    (void)n_in; (void)out_size; (void)d_ws; (void)ws_size;
    const float* xj   = (const float*)d_in[0];
    const float* W0   = (const float*)d_in[1];
    const float* Ws   = (const float*)d_in[2];
    const float* Wout = (const float*)d_in[3];
    float* out = (float*)d_out;

    const int n   = in_sizes[0] / (D_ + 1);   // 4194304
    constexpr int SPT = 8;                    // samples per thread
    constexpr int BLK = 256;                  // 8 wave32s per block

    if (n % (SPT * BLK) == 0) {
        // exact tiling (holds for n = 4194304): no per-iteration guards
        const int total_threads = n / SPT;
        const int blocks = total_threads / BLK;
        nnflow_kernel<SPT, false><<<blocks, BLK, 0, stream>>>(
            xj, W0, Ws, Wout, out, n, total_threads);
    } else {
        int total_threads = (n + SPT - 1) / SPT;
        int blocks = (total_threads + BLK - 1) / BLK;
        total_threads = blocks * BLK;
        nnflow_kernel<SPT, true><<<blocks, BLK, 0, stream>>>(
            xj, W0, Ws, Wout, out, n, total_threads);
    }
}